// MultiConvexNet_34583076667993
// MI455X (gfx1250) — compile-verified
//
#include <hip/hip_runtime.h>
#include <math.h>

typedef __attribute__((ext_vector_type(2))) float v2f;
typedef __attribute__((ext_vector_type(8))) float v8f;

#define N_DIRS 1986
#define NV 64
#define PPAIRS 16
#define TILES 125          // ceil(1986/16)
#define TG 16              // 16 tile-groups of 8 waves -> covers 125 tiles
#define FLB 1e-20f
#define FUB 1e20f
#define LOG2_10F 3.3219280948873623f

// flat output offsets (elements), reference return order
#define OFF_POINTS 0          // (8, 16*1986, 3)   = 762624
#define OFF_DIRH   762624     // (8, 16, 1986, 4)  = 1016832
#define OFF_OVER   1779456    // (8, 16, 1)        = 128
#define OFF_MEAN   1779584    // (8, 16, 3)        = 384
#define OFF_DIRS   1779968    // (1986, 3)         = 5958
#define OFF_LOCALV 1785926    // (8, 16, 64, 3)    = 24576

// ---------------------------------------------------------------------------
// Kernel 1: directions (double precision to match numpy f64 -> f32 cast) and
// zero the `overlap` output. One block.
// ---------------------------------------------------------------------------
__global__ void mcn_init_kernel(float* __restrict__ out) {
    const double PI = 3.14159265358979323846;
    const double STEP = PI / 32.0;   // linspace step for th1 (33 pts) and th2 (65 pts)
    for (int d = threadIdx.x; d < N_DIRS; d += blockDim.x) {
        double th1, th2;
        if (d < 1984) {
            int i = d / 64 + 1;      // i in 1..31
            int j = d % 64;          // j in 0..63
            th1 = -PI / 2.0 + STEP * (double)i;
            th2 = -PI + STEP * (double)j;
        } else {                     // poles: i=0 and i=32, j=0
            int i = (d == 1984) ? 0 : 32;
            th1 = -PI / 2.0 + STEP * (double)i;
            th2 = -PI;
        }
        double c1 = cos(th1), s1 = sin(th1);
        double c2 = cos(th2), s2 = sin(th2);
        out[OFF_DIRS + 3 * d + 0] = (float)(c1 * c2);
        out[OFF_DIRS + 3 * d + 1] = (float)(c1 * s2);
        out[OFF_DIRS + 3 * d + 2] = (float)(s1);
    }
    if (threadIdx.x < 128) out[OFF_OVER + threadIdx.x] = 0.0f;
}

// ---------------------------------------------------------------------------
// Kernel 2: fused smooth-projection transform.
// grid = 8 * 16 * TG blocks, 256 threads (8 waves). Each wave owns one
// 16-direction tile of one (b,p) pair. z-projection via V_WMMA_F32_16X16X4_F32.
// local_v lives in LDS as float4-interleaved [v*4 + c] (c==3 zero pad) so the
// WMMA A operand is one unconditional ds_load_b64 per tile.
// ---------------------------------------------------------------------------
__global__ __launch_bounds__(256)
void mcn_spt_kernel(const float* __restrict__ vertices,
                    const float* __restrict__ smoothness,
                    float* __restrict__ out) {
    __shared__ __align__(16) float lv4[NV * 4];
    __shared__ float meanv[3];

    const int blk = blockIdx.x;
    const int tg  = blk % TG;
    const int p   = (blk / TG) % PPAIRS;
    const int b   = blk / (TG * PPAIRS);
    const int tid = threadIdx.x;

    // ---- stage vertices, compute mean, center (whole block) ----
    const float* vb = vertices + (size_t)((b * PPAIRS + p) * NV) * 3;
    if (tid < NV) {
        lv4[tid * 4 + 0] = vb[tid * 3 + 0];
        lv4[tid * 4 + 1] = vb[tid * 3 + 1];
        lv4[tid * 4 + 2] = vb[tid * 3 + 2];
        lv4[tid * 4 + 3] = 0.0f;
    }
    __syncthreads();
    if (tid < 3) {
        float s = 0.0f;
        for (int v = 0; v < NV; ++v) s += lv4[v * 4 + tid];
        meanv[tid] = s * (1.0f / (float)NV);
    }
    __syncthreads();
    if (tid < NV) {
        lv4[tid * 4 + 0] -= meanv[0];
        lv4[tid * 4 + 1] -= meanv[1];
        lv4[tid * 4 + 2] -= meanv[2];
    }
    __syncthreads();

    if (tg == 0) {  // one block per (b,p) emits mean / local_v outputs
        if (tid < NV) {
            float* lo = out + OFF_LOCALV + (size_t)((b * PPAIRS + p) * NV + tid) * 3;
            lo[0] = lv4[tid * 4 + 0];
            lo[1] = lv4[tid * 4 + 1];
            lo[2] = lv4[tid * 4 + 2];
        }
        if (tid == 0) {
            float* mo = out + OFF_MEAN + (size_t)(b * PPAIRS + p) * 3;
            mo[0] = meanv[0]; mo[1] = meanv[1]; mo[2] = meanv[2];
        }
    }

    const int wave = tid >> 5;
    const int lane = tid & 31;
    const int tile = tg * 8 + wave;
    if (tile >= TILES) return;                 // wave-uniform exit

    const int half = lane >> 4;                // 0: K=0,1 half | 1: K=2,3 half
    const int col  = lane & 15;                // N index (direction in tile)
    const int d    = tile * 16 + col;
    const bool dvalid = (d < N_DIRS);

    const float* dirs = out + OFF_DIRS;        // produced by mcn_init_kernel
    float dx = 0.0f, dy = 0.0f, dz = 0.0f;
    if (dvalid) { dx = dirs[3 * d]; dy = dirs[3 * d + 1]; dz = dirs[3 * d + 2]; }

    // B (K=4 x N=16): lanes 0-15 carry K=0,1 ; lanes 16-31 carry K=2,3 (pad)
    v2f bop;
    bop.x = half ? dz : dx;
    bop.y = half ? 0.0f : dy;

    // z[v, d] for all 64 vertices: 4 WMMAs over 16-vertex A tiles (M=16,K=4).
    // A layout: lane -> M = lane%16, VGPRs hold K = 2*half, 2*half+1
    //   => one ds_load_b64 from lv4 at element (vrow*4 + 2*half).
    v8f acc[4];
#pragma unroll
    for (int t = 0; t < 4; ++t) {
        const int vrow = 16 * t + col;
        v2f aop = *(const v2f*)(lv4 + vrow * 4 + 2 * half);
        v8f c = {};
        acc[t] = __builtin_amdgcn_wmma_f32_16x16x4_f32(
            /*neg_a=*/false, aop, /*neg_b=*/false, bop,
            /*c_mod=*/(short)0, c, /*reuse_a=*/false, /*reuse_b=*/false);
    }
    // C layout: lane holds column d, rows v = 16*t + 8*half + r (r = 0..7);
    // the partner lane (lane ^ 16) holds the complementary rows.

    const float sp    = smoothness[b * PPAIRS + p];
    const float spm1  = sp - 1.0f;
    const float invsp = 1.0f / sp;

    // ---- per-element log2 of relu(z); max over v of relu(z) ----
    float lz[32];
    float zmax = 0.0f;
#pragma unroll
    for (int t = 0; t < 4; ++t)
#pragma unroll
        for (int r = 0; r < 8; ++r) {
            float zm = fmaxf(acc[t][r], 0.0f);
            zmax = fmaxf(zmax, zm);
            lz[t * 8 + r] = log2f(zm);         // -inf when zm == 0
        }
    zmax = fmaxf(zmax, __shfl_xor(zmax, 16, 32));

    // ---- dynamic-range scale k = 10^clip(ceil(lk), 0, 20) ----
    float zlog = log10f(zmax);                 // -inf when zmax == 0
    float expo = zlog * sp;
    float lk   = (expo < -20.0f) ? (-20.0f - expo) * invsp : 0.0f;
    lk = fminf(fmaxf(ceilf(lk), 0.0f), 20.0f); // ceil(inf)->inf -> 20
    const float k    = exp10f(lk);
    const float lg_k = lk * LOG2_10F;          // log2(k)

    // ---- sum of clip(zms^p) over v :  zms^p = exp2(p*(lz + lg_k)) ----
    const float pbase = sp * lg_k;
    float s = 0.0f;
#pragma unroll
    for (int t = 0; t < 4; ++t)
#pragma unroll
        for (int r = 0; r < 8; ++r) {
            const float l = lz[t * 8 + r];
            if (l > -INFINITY) {               // zm > 0
                float zp = exp2f(fmaf(l, sp, pbase));
                s += fminf(fmaxf(zp, FLB), FUB);
            }
        }
    s += __shfl_xor(s, 16, 32);
    float h = exp2f(log2f(s) * invsp);         // s^(1/p); s==0 -> 0
    h = fminf(fmaxf(h, FLB), FUB);
    const float lg_h = log2f(h);

    // ---- surf[d,:] = sum_v clip((zms/h)^(p-1)) * local_v[v,:] ----
    const float dbase = spm1 * (lg_k - lg_h);
    float sx = 0.0f, sy = 0.0f, sz = 0.0f;
#pragma unroll
    for (int t = 0; t < 4; ++t)
#pragma unroll
        for (int r = 0; r < 8; ++r) {
            const int v   = 16 * t + 8 * half + r;
            const float l = lz[t * 8 + r];
            float dh;
            if (l > -INFINITY) {
                dh = exp2f(fmaf(l, spm1, dbase));
                dh = fminf(fmaxf(dh, FLB), FUB);
            } else {
                dh = FLB;
            }
            const float4 lv = *(const float4*)(lv4 + v * 4);  // ds_load_b128
            sx += dh * lv.x;
            sy += dh * lv.y;
            sz += dh * lv.z;
        }
    sx += __shfl_xor(sx, 16, 32);
    sy += __shfl_xor(sy, 16, 32);
    sz += __shfl_xor(sz, 16, 32);

    // ---- emit (lanes 0-15 write; partner half has identical values) ----
    if (half == 0 && dvalid) {
        float hout = fminf(fmaxf(h / k, -FUB), FUB);
        float* pt = out + OFF_POINTS + (size_t)((b * PPAIRS + p) * N_DIRS + d) * 3;
        pt[0] = sx + meanv[0];
        pt[1] = sy + meanv[1];
        pt[2] = sz + meanv[2];
        float* dh4 = out + OFF_DIRH + (size_t)((b * PPAIRS + p) * N_DIRS + d) * 4;
        dh4[0] = dx; dh4[1] = dy; dh4[2] = dz; dh4[3] = hout;
    }
}

extern "C" void kernel_launch(void* const* d_in, const int* in_sizes, int n_in,
                              void* d_out, int out_size, void* d_ws, size_t ws_size,
                              hipStream_t stream) {
    const float* vertices   = (const float*)d_in[0];   // (8,16,64,3) f32
    const float* smoothness = (const float*)d_in[1];   // (8,16) f32
    float* out = (float*)d_out;

    hipLaunchKernelGGL(mcn_init_kernel, dim3(1), dim3(256), 0, stream, out);
    hipLaunchKernelGGL(mcn_spt_kernel, dim3(8 * PPAIRS * TG), dim3(256), 0, stream,
                       vertices, smoothness, out);
}